// LengthRegulator_65987877536065
// MI455X (gfx1250) — compile-verified
//
#include <hip/hip_runtime.h>

// LengthRegulator for MI455X (gfx1250, wave32).
// Bandwidth-bound op (~200MB HBM traffic -> ~9us @ 23.3 TB/s). Three kernels:
//  1) integrate(): per-(b,i) masked de-meaned prefix sum over Ty=4096, done as a
//     WMMA blocked scan: P = L16 x M via 4 chained V_WMMA_F32_16X16X4_F32
//     (fp32 in/out, numerically equivalent to a scalar fp32 scan).
//  2) flow composition: 8 Euler steps with linear-interp gathers (L2-resident field).
//  3) Catmull-Rom cubic gather, float4-coalesced streaming (the bandwidth kernel).

#define B_  16
#define TX_ 512
#define TY_ 4096
#define D_  512
#define NI_ 8

typedef __attribute__((ext_vector_type(2))) float v2f;
typedef __attribute__((ext_vector_type(8))) float v8f;

// ---------------------------------------------------------------------------
// Kernel 1: one wave32 per (b,i) chain. Shift/mask/de-mean, then inclusive
// prefix sum via WMMA over 16 tiles of 256 elements each.
//
// Tile layout: flat element (tile*256 + c*16 + r) lives at matrix (r,c).
// P = L x M (L = 16x16 lower-triangular ones) gives per-column inclusive
// prefixes; column carries via wave shuffles; tile carry kept in a register.
//
// fp32 A-matrix 16x4 VGPR layout (ISA 7.12.2): lanes 0-15 hold M=lane,
// K={0,1}; lanes 16-31 hold M=lane-16, K={2,3}. B 4x16 mirrors this with
// N striped across lanes. C/D: VGPR v -> M=v (lanes 0-15) / M=v+8 (16-31).
// ---------------------------------------------------------------------------
__global__ __launch_bounds__(32) void k_integrate(const float* __restrict__ ws,
                                                  const int* __restrict__ y_len,
                                                  float* __restrict__ wf) {
  const int chain = blockIdx.x;          // b*NI + i
  const int b     = chain / NI_;
  const int lane  = threadIdx.x;
  const int yl    = y_len[b];
  const float* src = ws + (size_t)chain * TY_;
  float*       dst = wf + (size_t)chain * TY_;

  // ---- pass 1: sum of shifted+masked sequence -> mean ----
  float s = 0.f;
  for (int t = lane; t < TY_; t += 32) {
    float v = (t >= 1 && t < yl) ? src[t - 1] : 0.f;
    s += v;
  }
#pragma unroll
  for (int off = 16; off > 0; off >>= 1) s += __shfl_xor(s, off, 32);
  const float mean     = s / (float)yl;
  const float outscale = 1.0f / (float)(NI_ * 4);

  __shared__ float lds[256];

  const int half = lane >> 4;   // 0: K low pair / M=0..7 rows ; 1: K high pair
  const int n    = lane & 15;   // column index (B/C/D) and row M (A)

  float carry = 0.f;
  for (int tile = 0; tile < TY_ / 256; ++tile) {
    const int tbase = tile * 256;

    // stage u[t] = v[t] - mean into LDS (flat order), coalesced global reads
#pragma unroll
    for (int k = 0; k < 8; ++k) {
      int t = tbase + k * 32 + lane;
      float v = (t >= 1 && t < yl) ? src[t - 1] : 0.f;
      lds[k * 32 + lane] = v - mean;
    }
    __syncthreads();

    // P = L16 x M : 4 chained K=4 fp32 WMMAs, fp32 accumulation in C
    v8f acc = {0.f, 0.f, 0.f, 0.f, 0.f, 0.f, 0.f, 0.f};
#pragma unroll
    for (int j = 0; j < 4; ++j) {
      const int kb = 4 * j + 2 * half;   // this lane's two K indices: kb, kb+1
      v2f a, bm;
      a.x  = (n >= kb)     ? 1.f : 0.f;  // L[m][k] = (k <= m)
      a.y  = (n >= kb + 1) ? 1.f : 0.f;
      bm.x = lds[n * 16 + kb];           // M[k][n] = u[n*16 + k]
      bm.y = lds[n * 16 + kb + 1];
      acc = __builtin_amdgcn_wmma_f32_16x16x4_f32(
          /*neg_a=*/false, a, /*neg_b=*/false, bm,
          /*c_mod=*/(short)0, acc, /*reuse_a=*/false, /*reuse_b=*/false);
    }
    __syncthreads();

    // column sums = row M=15 = acc[7] on lanes 16..31
    float colsum = acc[7];
    float inc = colsum;                   // inclusive scan across columns
#pragma unroll
    for (int off = 1; off < 16; off <<= 1) {
      float t = __shfl_up(inc, off, 32);
      if (half == 1 && n >= off) inc += t;
    }
    float excl     = inc - colsum;               // exclusive column prefix
    float tilesum  = __shfl(inc, 31, 32);        // total of this 256-tile
    float colcarry = __shfl(excl, 16 + n, 32);   // broadcast per column N
    const float addv = colcarry + carry;

    // scatter: (M,N) -> flat t = tbase + N*16 + M ; lane covers M = half*8+v
    const int tlane = tbase + n * 16 + half * 8;
#pragma unroll
    for (int v = 0; v < 8; ++v) {
      int t = tlane + v;
      float val = acc[v] + addv;
      dst[t] = (t < yl) ? val * outscale : 0.f;  // final mask + 1/(n_iter*4)
    }
    carry += tilesum;
  }
}

// ---------------------------------------------------------------------------
// Kernel 2: f = arange(Ty); 8x Euler steps f += linterp(w_i, f); scale by
// x_len/y_len. One thread per (b,ty). Gathers hit the 2MB L2-resident field.
// ---------------------------------------------------------------------------
__global__ __launch_bounds__(256) void k_flow(const float* __restrict__ wf,
                                              const int* __restrict__ x_len,
                                              const int* __restrict__ y_len,
                                              float* __restrict__ f_out) {
  const int idx = blockIdx.x * blockDim.x + threadIdx.x;   // b*TY + t
  const int b = idx >> 12;                                  // TY_ == 4096
  const int t = idx & (TY_ - 1);
  float f = (float)t;
#pragma unroll
  for (int i = 0; i < NI_; ++i) {
    const float* w = wf + (size_t)(b * NI_ + i) * TY_;
    float p  = fminf(fmaxf(f, 0.f), (float)(TY_ - 1));
    float fi = floorf(p);
    int   i0 = (int)fi;
    int   i1 = min(i0 + 1, TY_ - 1);
    float fr = p - fi;
    float w0 = w[i0];
    f += w0 + fr * (w[i1] - w0);
  }
  f *= (float)x_len[b] / (float)y_len[b];
  f_out[idx] = f;
}

// ---------------------------------------------------------------------------
// Kernel 3: Catmull-Rom cubic gather. One block per (b,ty); 128 threads x
// float4 cover D=512. 4 coalesced 2KB row reads (L2-served) + 2KB store.
// ---------------------------------------------------------------------------
__global__ __launch_bounds__(128) void k_cubic(const float* __restrict__ xs,
                                               const float* __restrict__ f,
                                               float* __restrict__ ys) {
  const int idx = blockIdx.x;              // b*TY + t
  const int b   = idx >> 12;

  const float fp = f[idx];
  float p  = fminf(fmaxf(fp, 0.f), (float)(TX_ - 1));
  float fi = floorf(p);
  int   i  = (int)fi;
  float t  = p - fi;
  float t2 = t * t, t3 = t2 * t;
  float w0 = -0.5f * t3 + t2 - 0.5f * t;
  float w1 =  1.5f * t3 - 2.5f * t2 + 1.0f;
  float w2 = -1.5f * t3 + 2.0f * t2 + 0.5f * t;
  float w3 =  0.5f * t3 - 0.5f * t2;

  const float* xb = xs + (size_t)b * TX_ * D_;
  const int r0 = max(i - 1, 0);
  const int r1 = i;                         // i already in [0, TX-1]
  const int r2 = min(i + 1, TX_ - 1);
  const int r3 = min(i + 2, TX_ - 1);
  const float4* g0 = (const float4*)(xb + (size_t)r0 * D_);
  const float4* g1 = (const float4*)(xb + (size_t)r1 * D_);
  const float4* g2 = (const float4*)(xb + (size_t)r2 * D_);
  const float4* g3 = (const float4*)(xb + (size_t)r3 * D_);

  const int d = threadIdx.x;                // float4 column index, 0..127
  float4 a = g0[d], c1 = g1[d], c2 = g2[d], c3 = g3[d];
  float4 o;
  o.x = w0 * a.x + w1 * c1.x + w2 * c2.x + w3 * c3.x;
  o.y = w0 * a.y + w1 * c1.y + w2 * c2.y + w3 * c3.y;
  o.z = w0 * a.z + w1 * c1.z + w2 * c2.z + w3 * c3.z;
  o.w = w0 * a.w + w1 * c1.w + w2 * c2.w + w3 * c3.w;
  ((float4*)(ys + (size_t)idx * D_))[d] = o;
}

extern "C" void kernel_launch(void* const* d_in, const int* in_sizes, int n_in,
                              void* d_out, int out_size, void* d_ws, size_t ws_size,
                              hipStream_t stream) {
  const float* xs = (const float*)d_in[0];   // (B, TX, D)
  const float* ws = (const float*)d_in[1];   // (B, NI, TY)
  const int*   xl = (const int*)d_in[2];     // (B,)
  const int*   yl = (const int*)d_in[3];     // (B,)

  float* ys = (float*)d_out;                               // (B, TY, D)
  float* f  = (float*)d_out + (size_t)B_ * TY_ * D_;       // (B, TY)
  float* wf = (float*)d_ws;                                // (B, NI, TY) scratch, 2MB

  k_integrate<<<B_ * NI_, 32, 0, stream>>>(ws, yl, wf);
  k_flow<<<(B_ * TY_) / 256, 256, 0, stream>>>(wf, xl, yl, f);
  k_cubic<<<B_ * TY_, 128, 0, stream>>>(xs, f, ys);
}